// GlobalTextFusion_22454089023933
// MI455X (gfx1250) — compile-verified
//
#include <hip/hip_runtime.h>
#include <hip/hip_bf16.h>

typedef __attribute__((ext_vector_type(2))) float v2f;
typedef __attribute__((ext_vector_type(8))) float v8f;

// ---------------------------------------------------------------------------
// Kernel 1: t = relu(BN(proj_w @ text_feat^T + proj_b))  via V_WMMA_F32_16X16X4_F32
// GEMM: M=256 (out ch, 16 tiles), N=16 (batch 8 padded), K=256.
// One wave (32 threads) per M-tile; K-loop of 64 steps of 4.
// A layout (16x4 f32): lane%16 = M row; VGPR0 = K={0|2}, VGPR1 = K={1|3} (hi half lanes +2)
// B layout (4x16 f32): lane%16 = N col; mirrored K split.
// D layout (16x16 f32, 8 VGPRs): c[v] = D[M = v + 8*(lane>>4)][N = lane%16]
// ---------------------------------------------------------------------------
__global__ __launch_bounds__(32) void k_proj_wmma(
    const float* __restrict__ proj_w, const float* __restrict__ text_feat,
    const float* __restrict__ proj_b, const float* __restrict__ bn_gamma,
    const float* __restrict__ bn_beta, const float* __restrict__ bn_mean,
    const float* __restrict__ bn_var, float* __restrict__ t_ws)
{
    const int C = 256;
    int lane = threadIdx.x;
    int row  = lane & 15;      // A: M row ; B/D: N col (batch)
    int hi   = lane >> 4;      // K half select
    int o_base = blockIdx.x * 16;

    int   nrow  = (row < 8) ? row : 0;
    float bmask = (row < 8) ? 1.0f : 0.0f;      // pad batch 8 -> 16 with zeros
    const float* aptr = proj_w   + (size_t)(o_base + row) * C + 2 * hi;
    const float* bptr = text_feat + (size_t)nrow * C + 2 * hi;

    v8f acc = {};
    for (int kb = 0; kb < C; kb += 4) {
        v2f a, b;
        a.x = aptr[kb];
        a.y = aptr[kb + 1];
        b.x = bptr[kb] * bmask;
        b.y = bptr[kb + 1] * bmask;
        // 8 args: (neg_a, A, neg_b, B, c_mod, C, reuse_a, reuse_b)
        acc = __builtin_amdgcn_wmma_f32_16x16x4_f32(false, a, false, b,
                                                    (short)0, acc, false, false);
    }

    if (row < 8) {
        int m0 = o_base + 8 * hi;
#pragma unroll
        for (int v = 0; v < 8; ++v) {
            int m = m0 + v;
            float raw = acc[v] + proj_b[m];
            float sc  = bn_gamma[m] * rsqrtf(bn_var[m] + 1e-5f);
            float t   = (raw - bn_mean[m]) * sc + bn_beta[m];
            t_ws[row * C + m] = fmaxf(t, 0.0f);
        }
    }
}

// ---------------------------------------------------------------------------
// Kernel 2: per-batch channel attention + avg/max stats. 1 block per batch.
// ---------------------------------------------------------------------------
__global__ __launch_bounds__(256) void k_attn(
    const float* __restrict__ t_ws,
    const float* __restrict__ ca_w1, const float* __restrict__ ca_b1,
    const float* __restrict__ ca_w2, const float* __restrict__ ca_b2,
    float* __restrict__ cw_ws, float* __restrict__ avg_ws, float* __restrict__ max_ws)
{
    const int C = 256, CR = 64;
    int b = blockIdx.x, tid = threadIdx.x;
    __shared__ float sh_h[CR];
    __shared__ float sh_sum[256];
    __shared__ float sh_max[256];

    const float* t = t_ws + b * C;
    float tv = t[tid];
    sh_sum[tid] = tv;
    sh_max[tid] = tv;

    if (tid < CR) {
        float a = ca_b1[tid];
        const float* w = ca_w1 + tid * C;
        for (int c = 0; c < C; ++c) a = fmaf(w[c], t[c], a);
        sh_h[tid] = fmaxf(a, 0.0f);
    }
    __syncthreads();

    {
        float a = ca_b2[tid];
        const float* w = ca_w2 + tid * CR;
#pragma unroll 8
        for (int r = 0; r < CR; ++r) a = fmaf(w[r], sh_h[r], a);
        cw_ws[b * C + tid] = 1.0f / (1.0f + __expf(-a));
    }

    for (int s = 128; s > 0; s >>= 1) {
        __syncthreads();
        if (tid < s) {
            sh_sum[tid] += sh_sum[tid + s];
            sh_max[tid] = fmaxf(sh_max[tid], sh_max[tid + s]);
        }
    }
    if (tid == 0) {
        avg_ws[b] = sh_sum[0] * (1.0f / 256.0f);
        max_ws[b] = sh_max[0];
    }
}

// ---------------------------------------------------------------------------
// Kernel 3: spatial-attention map. Conv input is spatially constant per batch,
// so the 7x7 zero-padded conv reduces to border-clipped kernel-window sums.
// ---------------------------------------------------------------------------
__global__ __launch_bounds__(256) void k_smap(
    const float* __restrict__ sa_w, const float* __restrict__ sa_b,
    const float* __restrict__ avg_ws, const float* __restrict__ max_ws,
    float* __restrict__ smap)
{
    const int H = 128, W = 128;
    __shared__ float w0[49], w1[49], bias;
    int tid = threadIdx.x;
    if (tid < 49)       w0[tid] = sa_w[tid];
    else if (tid < 98)  w1[tid - 49] = sa_w[tid];
    else if (tid == 98) bias = sa_b[0];
    __syncthreads();

    int idx = blockIdx.x * 256 + tid;   // over 8*128*128
    int b = idx >> 14;
    int p = idx & 16383;
    int y = p >> 7, x = p & 127;
    float av = avg_ws[b], mx = max_ws[b];
    float s0 = 0.0f, s1 = 0.0f;
#pragma unroll
    for (int ky = 0; ky < 7; ++ky) {
        int yy = y + ky - 3;
        if ((unsigned)yy >= (unsigned)H) continue;
#pragma unroll
        for (int kx = 0; kx < 7; ++kx) {
            int xx = x + kx - 3;
            if ((unsigned)xx >= (unsigned)W) continue;
            s0 += w0[ky * 7 + kx];
            s1 += w1[ky * 7 + kx];
        }
    }
    float sw = av * s0 + mx * s1 + bias;
    smap[idx] = 1.0f / (1.0f + __expf(-sw));
}

// ---------------------------------------------------------------------------
// Kernel 4: bulk streaming pass (bandwidth bound, ~270 MB -> ~11.6 us):
// out[b,c,y,x] = x * cw[b,c] * smap[b,y,x], float4 (b128) vectorized.
// 2^23 float4s total; plane = 4096 float4s, C = 256, B = 8 (all pow2).
// ---------------------------------------------------------------------------
__global__ __launch_bounds__(256) void k_fuse(
    const float4* __restrict__ x4, const float* __restrict__ cw_ws,
    const float4* __restrict__ sm4, float4* __restrict__ out4)
{
    unsigned i = blockIdx.x * 256u + threadIdx.x;   // < 2^23
    unsigned p4 = i & 4095u;          // float4 index within HxW plane
    unsigned c  = (i >> 12) & 255u;
    unsigned b  = i >> 20;
    float cwv = cw_ws[(b << 8) + c];
    float4 xv = x4[i];
    float4 sv = sm4[(b << 12) + p4];
    float4 o;
    o.x = xv.x * cwv * sv.x;
    o.y = xv.y * cwv * sv.y;
    o.z = xv.z * cwv * sv.z;
    o.w = xv.w * cwv * sv.w;
    out4[i] = o;
}

extern "C" void kernel_launch(void* const* d_in, const int* in_sizes, int n_in,
                              void* d_out, int out_size, void* d_ws, size_t ws_size,
                              hipStream_t stream) {
    const float* x         = (const float*)d_in[0];
    const float* text_feat = (const float*)d_in[1];
    const float* proj_w    = (const float*)d_in[2];
    const float* proj_b    = (const float*)d_in[3];
    const float* bn_gamma  = (const float*)d_in[4];
    const float* bn_beta   = (const float*)d_in[5];
    const float* bn_mean   = (const float*)d_in[6];
    const float* bn_var    = (const float*)d_in[7];
    const float* ca_w1     = (const float*)d_in[8];
    const float* ca_b1     = (const float*)d_in[9];
    const float* ca_w2     = (const float*)d_in[10];
    const float* ca_b2     = (const float*)d_in[11];
    const float* sa_w      = (const float*)d_in[12];
    const float* sa_b      = (const float*)d_in[13];
    float* out = (float*)d_out;

    float* ws     = (float*)d_ws;
    float* t_ws   = ws;            // 8*256
    float* cw_ws  = ws + 2048;     // 8*256
    float* avg_ws = ws + 4096;     // 8
    float* max_ws = ws + 4104;     // 8
    float* smap   = ws + 4608;     // 8*128*128, 16B aligned

    // 1) WMMA projection + BN + ReLU -> t (8,256)
    k_proj_wmma<<<dim3(16), dim3(32), 0, stream>>>(
        proj_w, text_feat, proj_b, bn_gamma, bn_beta, bn_mean, bn_var, t_ws);

    // 2) channel attention weights + per-batch avg/max
    k_attn<<<dim3(8), dim3(256), 0, stream>>>(
        t_ws, ca_w1, ca_b1, ca_w2, ca_b2, cw_ws, avg_ws, max_ws);

    // 3) spatial attention map (8,128,128)
    k_smap<<<dim3((8 * 128 * 128) / 256), dim3(256), 0, stream>>>(
        sa_w, sa_b, avg_ws, max_ws, smap);

    // 4) streaming fused scale: out = x * cw * smap
    k_fuse<<<dim3((8u * 256u * 128u * 128u / 4u) / 256u), dim3(256), 0, stream>>>(
        (const float4*)x, cw_ws, (const float4*)smap, (float4*)out);
}